// SimpleGCN_48704929137095
// MI455X (gfx1250) — compile-verified
//
#include <hip/hip_runtime.h>

// ---------------------------------------------------------------------------
// GCN forward for MI455X (gfx1250, wave32).
// Roofline: edge gather/scatter dominates (~1GB random traffic/layer) and the
// gathered xl (64MB padded) + scatter target agg (64MB) both fit in the 192MB
// L2 -> keep them resident, use HW global_atomic_add_f32. Node GEMMs are fp32
// -> V_WMMA_F32_16X16X4_F32 (no precision loss, GEMM is not the bottleneck).
// ---------------------------------------------------------------------------

#define NN      100000          // nodes  (== 6250 * 16, exact WMMA row tiles)
#define NE      800000
#define HID     150
#define DP      160             // padded hidden: 10 col tiles of 16, K%4==0
#define EF      7
#define NL      4
#define NG      128
#define ODIM    6
#define BN_EPS  1e-5f

typedef __attribute__((ext_vector_type(2))) float v2f;
typedef __attribute__((ext_vector_type(8))) float v8f;

// hardware fp32 atomic add (avoid CAS loops on the scatter path)
__device__ __forceinline__ void atomAddF(float* p, float v) {
#if defined(__has_builtin) && __has_builtin(__builtin_amdgcn_global_atomic_fadd_f32)
  typedef __attribute__((address_space(1))) float gfloat;
  __builtin_amdgcn_global_atomic_fadd_f32((gfloat*)(void*)p, v);
#else
  atomicAdd(p, v);
#endif
}

// ---------------- init / degree / norm ----------------
__global__ void k_deg_init(float* deg) {
  int n = blockIdx.x * blockDim.x + threadIdx.x;
  if (n < NN) deg[n] = 1.0f;
}
__global__ void k_deg_count(const int* __restrict__ row, float* deg) {
  int e = blockIdx.x * blockDim.x + threadIdx.x;
  if (e < NE) atomAddF(&deg[row[e]], 1.0f);
}
__global__ void k_dis(const float* __restrict__ deg, float* dis) {
  int n = blockIdx.x * blockDim.x + threadIdx.x;
  if (n < NN) dis[n] = rsqrtf(deg[n]);
}
__global__ void k_norm(const int* __restrict__ row, const int* __restrict__ col,
                       const float* __restrict__ dis, float* norm) {
  int e = blockIdx.x * blockDim.x + threadIdx.x;
  if (e < NE) norm[e] = dis[row[e]] * dis[col[e]];
}
__global__ void k_h_init(const int* __restrict__ x, const float* __restrict__ emb,
                         float* __restrict__ h) {
  int tid = blockIdx.x * blockDim.x + threadIdx.x;   // tid = n*DP + c
  if (tid >= NN * DP) return;
  int c = tid % DP, n = tid / DP;
  h[tid] = (c < HID) ? emb[x[n] * HID + c] : 0.0f;
}

// ---------------- weight padding for WMMA GEMM ----------------
__global__ void k_pad_w(const float* __restrict__ W, float* __restrict__ Wp) {
  int tid = blockIdx.x * blockDim.x + threadIdx.x;   // tid = k*DP + n
  if (tid >= DP * DP) return;
  int n = tid % DP, k = tid / DP;
  Wp[tid] = (k < HID && n < HID) ? W[k * HID + n] : 0.0f;
}

// ---------------- xl = h @ Wl + bl  via V_WMMA_F32_16X16X4_F32 ----------------
// One wave computes a 16-row strip across all 10 column tiles (A reused 10x).
// f32 A 16x4 layout: lane L holds row M=L&15, K = 2*(L>>4) + {0,1}  (v2f)
// f32 B 4x16 layout: vgpr r, lane L: K = r + 2*(L>>4), N = L&15     (v2f)
// f32 C/D 16x16:     vgpr r, lane L: M = r + 8*(L>>4), N = L&15     (v8f)
__global__ void __launch_bounds__(256)
k_gemm_xl(const float* __restrict__ h, const float* __restrict__ Wp,
          const float* __restrict__ bl, float* __restrict__ xl) {
  int wave = (blockIdx.x * blockDim.x + threadIdx.x) >> 5;
  int lane = threadIdx.x & 31;
  int rowBase = wave * 16;
  if (rowBase >= NN) return;              // wave-uniform exit: EXEC stays all-1s
  int m  = lane & 15;
  int kh = lane >> 4;                     // 0 or 1

  v8f acc[10] = {};
  const float* arow = h + (size_t)(rowBase + m) * DP + 2 * kh;

  for (int k = 0; k < DP; k += 4) {
    v2f a;
    a.x = arow[k];
    a.y = arow[k + 1];
    const float* b0 = Wp + (size_t)(k + 2 * kh) * DP + m;
    #pragma unroll
    for (int t = 0; t < 10; ++t) {
      v2f b;
      b.x = b0[t * 16];                   // K = 2*kh
      b.y = b0[t * 16 + DP];              // K = 2*kh + 1
      acc[t] = __builtin_amdgcn_wmma_f32_16x16x4_f32(
          false, a, false, b, (short)0, acc[t], false, false);
    }
  }

  #pragma unroll
  for (int t = 0; t < 10; ++t) {
    #pragma unroll
    for (int r = 0; r < 8; ++r) {
      int rowo = rowBase + r + 8 * kh;
      int colo = t * 16 + m;
      float v = acc[t][r] + (colo < HID ? bl[colo] : 0.0f);
      xl[(size_t)rowo * DP + colo] = v;
    }
  }
}

// ---------------- fused edge encoder + message + scatter-add ----------------
// msg = norm[e] * relu(xl[row] + edge_attr@We + be); agg[col] += msg
__global__ void __launch_bounds__(256)
k_edge_msg(const int* __restrict__ row, const int* __restrict__ col,
           const float* __restrict__ eattr, const float* __restrict__ We,
           const float* __restrict__ be, const float* __restrict__ norm,
           const float* __restrict__ xl, float* __restrict__ agg) {
  __shared__ float sWe[EF * DP];
  __shared__ float sbe[DP];
  for (int i = threadIdx.x; i < EF * HID; i += blockDim.x)
    sWe[(i / HID) * DP + (i % HID)] = We[i];
  for (int i = threadIdx.x; i < HID; i += blockDim.x) sbe[i] = be[i];
  __syncthreads();

  const int EPW = 8;                       // edges per wave
  int wave = threadIdx.x >> 5;
  int lane = threadIdx.x & 31;
  int eBase = (blockIdx.x * 8 + wave) * EPW;

  for (int ei = 0; ei < EPW; ++ei) {
    int e = eBase + ei;
    if (e >= NE) return;
    int   rs = row[e], cs = col[e];
    float nm = norm[e];
    float ea[EF];
    #pragma unroll
    for (int f = 0; f < EF; ++f) ea[f] = eattr[(size_t)e * EF + f];
    const float* xr = xl + (size_t)rs * DP;
    float* ag = agg + (size_t)cs * DP;
    for (int d = lane; d < HID; d += 32) {
      float enc = sbe[d];
      #pragma unroll
      for (int f = 0; f < EF; ++f) enc = fmaf(ea[f], sWe[f * DP + d], enc);
      float v = fmaxf(xr[d] + enc, 0.0f) * nm;
      atomAddF(ag + d, v);
    }
  }
}

// ---------------- out = agg + relu(xl+root)/deg ; accumulate BN stats --------
__global__ void k_combine_stats(const float* __restrict__ xl, const float* __restrict__ deg,
                                const float* __restrict__ root, float* __restrict__ out,
                                float* __restrict__ sums, float* __restrict__ sumsq) {
  int c = threadIdx.x;                     // blockDim = 160 (5 waves)
  int n0 = blockIdx.x * 256;
  int n1 = n0 + 256; if (n1 > NN) n1 = NN;
  if (c >= HID) return;
  float rt = root[c], s = 0.0f, s2 = 0.0f;
  for (int n = n0; n < n1; ++n) {
    size_t i = (size_t)n * DP + c;
    float v = out[i] + fmaxf(xl[i] + rt, 0.0f) / deg[n];
    out[i] = v;
    s += v; s2 += v * v;
  }
  atomAddF(&sums[c], s);
  atomAddF(&sumsq[c], s2);
}

__global__ void k_bn_param(const float* __restrict__ sums, const float* __restrict__ sumsq,
                           float* __restrict__ mu, float* __restrict__ rsig) {
  int c = threadIdx.x;
  if (c >= HID) return;
  const float invN = 1.0f / (float)NN;
  float m = sums[c] * invN;
  float var = sumsq[c] * invN - m * m;
  mu[c] = m;
  rsig[c] = rsqrtf(var + BN_EPS);
}

// ---------------- h += relu?(gamma*(out-mu)*rsig + beta)  (residual) --------
__global__ void k_bn_apply(const float* __restrict__ out, const float* __restrict__ mu,
                           const float* __restrict__ rsig, const float* __restrict__ gamma,
                           const float* __restrict__ beta, float* __restrict__ h, int doRelu) {
  int tid = blockIdx.x * blockDim.x + threadIdx.x;   // tid = n*DP + c
  if (tid >= NN * DP) return;
  int c = tid % DP;
  if (c >= HID) return;
  float v = gamma[c] * (out[tid] - mu[c]) * rsig[c] + beta[c];
  if (doRelu) v = fmaxf(v, 0.0f);
  h[tid] += v;
}

// ---------------- global mean pool + final projection ----------------
__global__ void k_pool_cnt(const int* __restrict__ batch, float* __restrict__ gcnt) {
  int n = blockIdx.x * blockDim.x + threadIdx.x;
  if (n < NN) atomAddF(&gcnt[batch[n]], 1.0f);
}
__global__ void k_pool_sum(const float* __restrict__ h, const int* __restrict__ batch,
                           float* __restrict__ gsum) {
  int tid = blockIdx.x * blockDim.x + threadIdx.x;   // tid = n*DP + c
  if (tid >= NN * DP) return;
  int c = tid % DP, n = tid / DP;
  if (c >= HID) return;
  atomAddF(&gsum[batch[n] * DP + c], h[tid]);
}
__global__ void k_final(const float* __restrict__ gsum, const float* __restrict__ gcnt,
                        const float* __restrict__ Wp, const float* __restrict__ bp,
                        float* __restrict__ outp) {
  int tid = blockIdx.x * blockDim.x + threadIdx.x;
  if (tid >= NG * ODIM) return;
  int g = tid / ODIM, o = tid % ODIM;
  float cnt = gcnt[g]; if (cnt < 1.0f) cnt = 1.0f;
  float inv = 1.0f / cnt;
  float acc = bp[o];
  for (int d = 0; d < HID; ++d)
    acc = fmaf(gsum[g * DP + d] * inv, Wp[d * ODIM + o], acc);
  outp[tid] = acc;
}

// ---------------------------------------------------------------------------
extern "C" void kernel_launch(void* const* d_in, const int* in_sizes, int n_in,
                              void* d_out, int out_size, void* d_ws, size_t ws_size,
                              hipStream_t stream) {
  const int*   x      = (const int*)  d_in[0];
  const int*   erow   = (const int*)  d_in[1];
  const int*   ecol   = erow + NE;
  const float* eattr  = (const float*)d_in[2];
  const int*   batch  = (const int*)  d_in[3];
  const float* emb    = (const float*)d_in[4];
  const float* Wl     = (const float*)d_in[5];
  const float* bl     = (const float*)d_in[6];
  const float* We     = (const float*)d_in[7];
  const float* be     = (const float*)d_in[8];
  const float* root   = (const float*)d_in[9];
  const float* gamma  = (const float*)d_in[10];
  const float* beta   = (const float*)d_in[11];
  const float* Wp     = (const float*)d_in[12];
  const float* bp     = (const float*)d_in[13];
  float* outp = (float*)d_out;

  // workspace layout (floats); total ~197 MB
  float* ws    = (float*)d_ws;
  float* h     = ws;                               // NN*DP
  float* xl    = h    + (size_t)NN * DP;           // NN*DP
  float* agg   = xl   + (size_t)NN * DP;           // NN*DP (also "out")
  float* deg   = agg  + (size_t)NN * DP;           // NN
  float* dis   = deg  + NN;                        // NN
  float* norm  = dis  + NN;                        // NE
  float* Wpad  = norm + NE;                        // DP*DP
  float* sums  = Wpad + DP * DP;                   // DP
  float* sumsq = sums + DP;                        // DP
  float* mu    = sumsq + DP;                       // DP
  float* rsig  = mu + DP;                          // DP
  float* gsum  = rsig + DP;                        // NG*DP
  float* gcnt  = gsum + NG * DP;                   // NG

  const int ND = NN * DP;                          // 16,000,000

  // degrees + norm (layer-independent)
  k_deg_init <<<(NN + 255) / 256, 256, 0, stream>>>(deg);
  k_deg_count<<<(NE + 255) / 256, 256, 0, stream>>>(erow, deg);
  k_dis      <<<(NN + 255) / 256, 256, 0, stream>>>(deg, dis);
  k_norm     <<<(NE + 255) / 256, 256, 0, stream>>>(erow, ecol, dis, norm);
  k_h_init   <<<(ND + 255) / 256, 256, 0, stream>>>(x, emb, h);

  for (int l = 0; l < NL; ++l) {
    k_pad_w<<<(DP * DP) / 256, 256, 0, stream>>>(Wl + (size_t)l * HID * HID, Wpad);
    k_gemm_xl<<<(NN / 16 + 7) / 8, 256, 0, stream>>>(h, Wpad, bl + l * HID, xl);
    hipMemsetAsync(agg, 0, (size_t)ND * sizeof(float), stream);
    hipMemsetAsync(sums, 0, 2 * DP * sizeof(float), stream);   // sums+sumsq
    k_edge_msg<<<NE / 64, 256, 0, stream>>>(erow, ecol, eattr,
                                            We + (size_t)l * EF * HID, be + l * HID,
                                            norm, xl, agg);
    k_combine_stats<<<(NN + 255) / 256, DP, 0, stream>>>(xl, deg, root + l * HID,
                                                         agg, sums, sumsq);
    k_bn_param<<<1, DP, 0, stream>>>(sums, sumsq, mu, rsig);
    k_bn_apply<<<ND / 256, 256, 0, stream>>>(agg, mu, rsig, gamma + l * HID,
                                             beta + l * HID, h, (l < NL - 1) ? 1 : 0);
  }

  hipMemsetAsync(gsum, 0, (size_t)(NG * DP + NG) * sizeof(float), stream);  // gsum+gcnt
  k_pool_cnt<<<(NN + 255) / 256, 256, 0, stream>>>(batch, gcnt);
  k_pool_sum<<<ND / 256, 256, 0, stream>>>(h, batch, gsum);
  k_final<<<(NG * ODIM + 255) / 256, 256, 0, stream>>>(gsum, gcnt, Wp, bp, outp);
}